// FullModel_55542517071921
// MI455X (gfx1250) — compile-verified
//
#include <hip/hip_runtime.h>
#include <math.h>

// Problem dims (fixed by the reference)
#define T_STEPS 256
#define NSAT    1024
#define GSAT    2048
#define HID     256
#define IN_DIM  32

// Fused GEMM tiling
#define KCH     32          // K-chunk staged in LDS
#define NCHUNK  9           // 1 chunk input-proj (K=32) + 8 chunks recurrent (K=256)
#define KPAD    34          // padded K-stride in LDS words (bank-conflict free, 8B aligned)

typedef __attribute__((ext_vector_type(2))) float v2f;
typedef __attribute__((ext_vector_type(8))) float v8f;

__device__ __forceinline__ float sigmoidf_(float x) {
    return 1.0f / (1.0f + __expf(-x));
}

// CDNA5 async global->LDS copy (GLOBAL_LOAD_ASYNC_TO_LDS_B32, ASYNCcnt-tracked)
__device__ __forceinline__ void async_ld_f32(const float* gptr, float* lptr) {
    unsigned int loff = (unsigned int)(size_t)lptr;   // LDS aperture: low 32 bits = LDS offset
    asm volatile("global_load_async_to_lds_b32 %0, %1, off"
                 :: "v"(loff), "v"(gptr) : "memory");
}
__device__ __forceinline__ void wait_async0() {
    asm volatile("s_wait_asynccnt 0x0" ::: "memory");
}

// ---------------------------------------------------------------------------
// gather_prep: blocks 0..255  -> Hsat[n] = Hg[sid[n]]     (A-operand snapshot)
//              blocks 256..259-> rvec[g*256+o] = r2s_lb + b_sat + h_rec . r2s_lw
//              block 260      -> mean_sat[o] = mean_n Hg[sid[n]][o]
// ---------------------------------------------------------------------------
__global__ void gather_prep_kernel(const float* __restrict__ Hg,
                                   const float* __restrict__ h_rec,
                                   const float* __restrict__ r2s_lw,
                                   const float* __restrict__ r2s_lb,
                                   const float* __restrict__ b_sat,
                                   const int*   __restrict__ sid,
                                   float* __restrict__ Hsat,
                                   float* __restrict__ rvec,
                                   float* __restrict__ mean_sat)
{
    const int o   = threadIdx.x;
    const int blk = blockIdx.x;
    if (blk < 256) {
        #pragma unroll
        for (int i = 0; i < 4; ++i) {
            const int n = blk * 4 + i;
            Hsat[(size_t)n * HID + o] = Hg[(size_t)sid[n] * HID + o];
        }
    } else if (blk < 260) {
        const int g = blk - 256;
        float acc = r2s_lb[g*HID + o] + b_sat[g*HID + o];
        const float* w = r2s_lw + (size_t)g*HID*HID + o;
        #pragma unroll 4
        for (int h = 0; h < HID; ++h) acc += h_rec[h] * w[h*HID];
        rvec[g*HID + o] = acc;
    } else {
        float acc = 0.0f;
        #pragma unroll 4
        for (int n = 0; n < NSAT; ++n) acc += Hg[(size_t)sid[n]*HID + o];
        mean_sat[o] = acc * (1.0f / (float)NSAT);
    }
}

// ---------------------------------------------------------------------------
// rec: receiver LSTM cell (one block, 256 threads, thread = hid index o)
// ---------------------------------------------------------------------------
__global__ void rec_kernel(const float* __restrict__ xr,       // [IN]
                           const float* __restrict__ W_rec,    // [4,IN,HID]
                           const float* __restrict__ b_rec,    // [4,HID]
                           const float* __restrict__ s2r_lw,   // [4,HID,HID]
                           const float* __restrict__ s2r_lb,   // [4,HID]
                           const float* __restrict__ s2r_rw,   // [4,HID,HID]
                           const float* __restrict__ mean_sat, // [HID]
                           const float* __restrict__ h_old,    // [HID]
                           const float* __restrict__ c_old,    // [HID]
                           float* __restrict__ h_new,
                           float* __restrict__ c_new)
{
    const int o = threadIdx.x;
    float pre[4];
    #pragma unroll
    for (int g = 0; g < 4; ++g) {
        float acc = b_rec[g*HID + o] + s2r_lb[g*HID + o];
        const float* w1 = W_rec + (size_t)g*IN_DIM*HID + o;
        #pragma unroll 4
        for (int i = 0; i < IN_DIM; ++i) acc += xr[i] * w1[i*HID];
        const float* w2 = s2r_lw + (size_t)g*HID*HID + o;
        const float* w3 = s2r_rw + (size_t)g*HID*HID + o;
        #pragma unroll 4
        for (int h = 0; h < HID; ++h)
            acc += mean_sat[h] * w2[h*HID] + h_old[h] * w3[h*HID];
        pre[g] = acc;
    }
    const float ig = sigmoidf_(pre[0]);
    const float fg = sigmoidf_(pre[1]);
    const float tg = tanhf(pre[2]);
    const float og = sigmoidf_(pre[3]);
    const float cn = fg * c_old[o] + ig * tg;
    c_new[o] = cn;
    h_new[o] = og * tanhf(cn);
}

// ---------------------------------------------------------------------------
// sat_fused: for all 1024 sats, all 4 gates:
//   pre[g][n][o] = xs[n].W_sat[g][:,o] + Hsat[n].r2s_rw[g][:,o] + rvec[g*256+o]
//   then LSTM gates + in-place scatter to Hg/Cg at row sid[n].
// Grid (16,8); 256 threads = 8 waves as 4(M) x 2(o-half).
// Block tile: 64 rows x 32 o x 4 gates. Wave tile: 16x16 per gate (4 v8f accs).
// B (both W_sat and r2s_rw sections) staged in LDS, transposed Bl[col][k],
// double-buffered per 32-K chunk via async global->LDS copies.
// ---------------------------------------------------------------------------
__global__ __launch_bounds__(256) void sat_fused_kernel(
    const float* __restrict__ xs,     // [NSAT, IN]  x_sat at step t
    const float* __restrict__ W_sat,  // [4, IN, HID]
    const float* __restrict__ Hsat,   // [NSAT, HID] gathered h_sat snapshot
    const float* __restrict__ R,      // r2s_rw [4, HID, HID]
    const float* __restrict__ rvec,   // [4*HID]
    const int*   __restrict__ sid,    // [NSAT]
    float* __restrict__ Hg,           // [GSAT, HID] scatter target
    float* __restrict__ Cg)           // [GSAT, HID] scatter target
{
    __shared__ float Bl[2][128 * KPAD];   // 2 x 17408 B

    const int tid    = threadIdx.x;
    const int wave   = tid >> 5;
    const int lane   = tid & 31;
    const int lane16 = lane & 15;
    const int lhalf  = lane >> 4;

    const int mwave  = wave >> 1;            // 0..3
    const int ohalf  = (wave & 1) * 16;      // 0 / 16
    const int m_base = blockIdx.x * 64 + mwave * 16;
    const int o_base = blockIdx.y * 32;

    // A rows: lanes 0-15 carry M=0..15, lane-half selects K-pair
    const int    row   = m_base + lane16;
    const float* arow1 = xs   + (size_t)row * IN_DIM;
    const float* arow2 = Hsat + (size_t)row * HID;

    // Staging mapping: 256 threads cover 128 cols (4 gates x 32 o) x 2 k-rows
    const int cc = tid & 127;                // col = gate*32 + o-offset
    const int sg = cc >> 5;                  // gate for staging
    const int so = o_base + (cc & 31);       // o column for staging
    const int k0 = tid >> 7;                 // 0..1
    float* lcol = &Bl[0][0] + cc * KPAD;

    // Accumulators: one 16x16 tile per gate, init with rvec broadcast
    v8f acc[4];
    #pragma unroll
    for (int g = 0; g < 4; ++g) {
        const float rv = rvec[g*HID + o_base + ohalf + lane16];
        acc[g] = (v8f){rv, rv, rv, rv, rv, rv, rv, rv};
    }

    // Stage chunk 0: input-projection weights W_sat rows k=0..31
    {
        const float* src = W_sat + (size_t)sg * IN_DIM * HID + so;
        #pragma unroll
        for (int k = k0; k < KCH; k += 2)
            async_ld_f32(src + (size_t)k * HID, lcol + k);
    }
    wait_async0();
    __syncthreads();

    for (int ch = 0; ch < NCHUNK; ++ch) {
        const int buf = ch & 1;

        // Prefetch next chunk into the other buffer (recurrent weights r2s_rw)
        if (ch + 1 < NCHUNK) {
            const float* src = R + (size_t)sg * HID * HID + (size_t)(ch * KCH) * HID + so;
            float* dst = lcol + (buf ^ 1) * (128 * KPAD);
            #pragma unroll
            for (int k = k0; k < KCH; k += 2)
                async_ld_f32(src + (size_t)k * HID, dst + k);
        }

        // Compute 8 k-steps (K=4 each) on current buffer
        const float* ar = (ch == 0) ? arow1 : (arow2 + (ch - 1) * KCH);
        const float* lb = &Bl[buf][0] + (ohalf + lane16) * KPAD;
        #pragma unroll
        for (int ks = 0; ks < KCH/4; ++ks) {
            const int kk = ks*4 + 2*lhalf;
            v2f a; a.x = ar[kk]; a.y = ar[kk+1];
            #pragma unroll
            for (int g = 0; g < 4; ++g) {
                const float* bp = lb + g * (32 * KPAD) + kk;
                v2f b; b.x = bp[0]; b.y = bp[1];
                acc[g] = __builtin_amdgcn_wmma_f32_16x16x4_f32(
                    false, a, false, b, (short)0, acc[g], false, false);
            }
        }

        wait_async0();
        __syncthreads();
    }

    // Fused LSTM epilogue + scatter (each (row,o) owned by exactly one lane)
    const int o = o_base + ohalf + lane16;
    #pragma unroll
    for (int r = 0; r < 8; ++r) {
        const int m = m_base + lhalf*8 + r;
        const size_t idx = (size_t)sid[m] * HID + o;
        const float ig = sigmoidf_(acc[0][r]);
        const float fg = sigmoidf_(acc[1][r]);
        const float tg = tanhf    (acc[2][r]);
        const float og = sigmoidf_(acc[3][r]);
        const float cn = fg * Cg[idx] + ig * tg;
        Cg[idx] = cn;
        Hg[idx] = og * tanhf(cn);
    }
}

// ---------------------------------------------------------------------------
// out: pred = h_rec @ W_out + b_out ; d_out = [pred(2), y_true(2)]
// ---------------------------------------------------------------------------
__global__ void out_kernel(const float* __restrict__ h_rec,
                           const float* __restrict__ W_out,  // [HID,2]
                           const float* __restrict__ b_out,  // [2]
                           const float* __restrict__ y_true, // [2]
                           float* __restrict__ out)
{
    const int j = threadIdx.x;
    if (j < 2) {
        float acc = b_out[j];
        for (int o = 0; o < HID; ++o) acc += h_rec[o] * W_out[o*2 + j];
        out[j]     = acc;
        out[2 + j] = y_true[j];
    }
}

// ---------------------------------------------------------------------------
extern "C" void kernel_launch(void* const* d_in, const int* in_sizes, int n_in,
                              void* d_out, int out_size, void* d_ws, size_t ws_size,
                              hipStream_t stream)
{
    (void)in_sizes; (void)n_in; (void)out_size; (void)ws_size;

    const float* x_rec  = (const float*)d_in[0];   // [T,1,IN]
    const float* x_sat  = (const float*)d_in[1];   // [T,NSAT,IN]
    const float* y_true = (const float*)d_in[2];   // [1,2]
    const float* W_rec  = (const float*)d_in[3];
    const float* b_rec  = (const float*)d_in[4];
    const float* W_sat  = (const float*)d_in[5];
    const float* b_sat  = (const float*)d_in[6];
    const float* s2r_lw = (const float*)d_in[7];
    const float* s2r_lb = (const float*)d_in[8];
    const float* s2r_rw = (const float*)d_in[9];
    const float* r2s_lw = (const float*)d_in[10];
    const float* r2s_lb = (const float*)d_in[11];
    const float* r2s_rw = (const float*)d_in[12];
    const float* W_out  = (const float*)d_in[13];
    const float* b_out  = (const float*)d_in[14];
    const int*   s_ids  = (const int*)d_in[15];    // [T,NSAT]

    // Workspace layout (floats)
    float* ws       = (float*)d_ws;
    float* Hg       = ws;                              // GSAT*HID
    float* Cg       = Hg   + (size_t)GSAT*HID;         // GSAT*HID
    float* Hsat     = Cg   + (size_t)GSAT*HID;         // NSAT*HID
    float* hbuf     = Hsat + (size_t)NSAT*HID;         // 2*HID (ping-pong)
    float* cbuf     = hbuf + 2*HID;                    // 2*HID (ping-pong)
    float* mean_sat = cbuf + 2*HID;                    // HID
    float* rvec     = mean_sat + HID;                  // 4*HID

    // Zero recurrent state (graph-capture safe)
    hipMemsetAsync(Hg,   0, (size_t)2*GSAT*HID*sizeof(float), stream); // Hg+Cg contiguous
    hipMemsetAsync(hbuf, 0, (size_t)4*HID*sizeof(float),      stream); // hbuf+cbuf contiguous

    for (int t = 0; t < T_STEPS; ++t) {
        const int cur = t & 1;
        const int nxt = cur ^ 1;
        const int* sid_t = s_ids + (size_t)t * NSAT;

        gather_prep_kernel<<<261, 256, 0, stream>>>(
            Hg, hbuf + cur*HID, r2s_lw, r2s_lb, b_sat, sid_t,
            Hsat, rvec, mean_sat);

        rec_kernel<<<1, 256, 0, stream>>>(
            x_rec + (size_t)t*IN_DIM, W_rec, b_rec, s2r_lw, s2r_lb, s2r_rw,
            mean_sat, hbuf + cur*HID, cbuf + cur*HID,
            hbuf + nxt*HID, cbuf + nxt*HID);

        sat_fused_kernel<<<dim3(16, 8), 256, 0, stream>>>(
            x_sat + (size_t)t*NSAT*IN_DIM, W_sat, Hsat, r2s_rw, rvec, sid_t,
            Hg, Cg);
    }

    // T even -> final h_rec lives in buffer 0
    out_kernel<<<1, 64, 0, stream>>>(hbuf, W_out, b_out, y_true, (float*)d_out);
}